// HybridQFCModel_65481071406840
// MI455X (gfx1250) — compile-verified
//
#include <hip/hip_runtime.h>
#include <hip/hip_bf16.h>
#include <math.h>

typedef float v2f __attribute__((ext_vector_type(2)));
typedef float v8f __attribute__((ext_vector_type(8)));

#define NW 4
#define DIM 16

// ---------------------------------------------------------------------------
// Kernel 0: compose the fixed (batch-independent) part of the circuit into a
// 16x16 complex unitary U = [Ur | Ui], stored row-major in workspace.
// 16 lanes, lane j owns column j; every gate acts within a column.
// ---------------------------------------------------------------------------
__global__ __launch_bounds__(32) void build_unitary(const float* __restrict__ W,
                                                    float* __restrict__ Ur,
                                                    float* __restrict__ Ui) {
    int j = threadIdx.x;
    if (j >= DIM) return;
    float vr[DIM], vi[DIM];
    for (int i = 0; i < DIM; ++i) { vr[i] = (i == j) ? 1.0f : 0.0f; vi[i] = 0.0f; }

    for (int l = 0; l < 2; ++l) {
        for (int w = 0; w < NW; ++w) {
            const float* g = W + (l * NW + w) * 3;
            const int mask = 1 << (3 - w);
            // RX(t): [[c, -i s], [-i s, c]]
            {
                float c = cosf(0.5f * g[0]), s = sinf(0.5f * g[0]);
                for (int i = 0; i < DIM; ++i) if (!(i & mask)) {
                    int i1 = i | mask;
                    float r0 = vr[i], m0 = vi[i], r1 = vr[i1], m1 = vi[i1];
                    vr[i]  = c * r0 + s * m1;  vi[i]  = c * m0 - s * r1;
                    vr[i1] = s * m0 + c * r1;  vi[i1] = -s * r0 + c * m1;
                }
            }
            // RY(t): [[c, -s], [s, c]] (real)
            {
                float c = cosf(0.5f * g[1]), s = sinf(0.5f * g[1]);
                for (int i = 0; i < DIM; ++i) if (!(i & mask)) {
                    int i1 = i | mask;
                    float r0 = vr[i], m0 = vi[i], r1 = vr[i1], m1 = vi[i1];
                    vr[i]  = c * r0 - s * r1;  vi[i]  = c * m0 - s * m1;
                    vr[i1] = s * r0 + c * r1;  vi[i1] = s * m0 + c * m1;
                }
            }
            // RZ(t): diag(e^{-it/2}, e^{+it/2})
            {
                float c = cosf(0.5f * g[2]), s = sinf(0.5f * g[2]);
                for (int i = 0; i < DIM; ++i) if (!(i & mask)) {
                    int i1 = i | mask;
                    float r0 = vr[i], m0 = vi[i], r1 = vr[i1], m1 = vi[i1];
                    vr[i]  = c * r0 + s * m0;  vi[i]  = c * m0 - s * r0;
                    vr[i1] = c * r1 - s * m1;  vi[i1] = c * m1 + s * r1;
                }
            }
        }
        // CNOT ring: control w, target (w+1)%4 ; new[i] = old[perm[i]]
        for (int w = 0; w < NW; ++w) {
            int cmask = 1 << (3 - w);
            int tmask = 1 << (3 - ((w + 1) & 3));
            float tr[DIM], ti[DIM];
            for (int i = 0; i < DIM; ++i) {
                int src = (i & cmask) ? (i ^ tmask) : i;
                tr[i] = vr[src]; ti[i] = vi[src];
            }
            for (int i = 0; i < DIM; ++i) { vr[i] = tr[i]; vi[i] = ti[i]; }
        }
    }
    for (int i = 0; i < DIM; ++i) { Ur[i * 16 + j] = vr[i]; Ui[i * 16 + j] = vi[i]; }
}

// Cross-half (lane XOR 16) add via DS_SWIZZLE SWAPX16 (wave32, EXEC all ones).
__device__ __forceinline__ float xor16_add(float v) {
    int sw = __builtin_amdgcn_ds_swizzle(__float_as_int(v), 0x401f); // xor=0x10,and=0x1f
    return v + __int_as_float(sw);
}

// ---------------------------------------------------------------------------
// Kernel 1: pool -> encode product state -> WMMA f32 16x16x4 complex matvec
// -> probs -> Z expectations -> store q + per-block batchnorm partials.
// 256 threads = 8 waves; 1 sample/lane; each wave does 2 tiles of 16 samples.
// ---------------------------------------------------------------------------
__global__ __launch_bounds__(256) void qcirc_main(const float* __restrict__ x,
                                                  const float* __restrict__ Ur,
                                                  const float* __restrict__ Ui,
                                                  float* __restrict__ out,
                                                  float* __restrict__ partials) {
    __shared__ float PsiRe[16][264];  // stride 264 -> conflict-free row pairs
    __shared__ float PsiIm[16][264];
    __shared__ float red[8][256];

    const int tid  = threadIdx.x;
    const int lane = tid & 31;
    const int wave = tid >> 5;
    const int row  = lane & 15;
    const int hi   = lane >> 4;
    const long sample = (long)blockIdx.x * 256 + tid;

    // ---- pooling: each lane streams its own sample's 144 floats (36 x b128)
    const float4* xp = (const float4*)(x + sample * 144);
    float acc[4] = {0.f, 0.f, 0.f, 0.f};
#pragma unroll
    for (int t = 0; t < 36; ++t) {
        float4 v = xp[t];
        const int r = t / 3, part = t % 3, rb = r / 6;  // constants after unroll
        if (part == 0)      { acc[2 * rb]     += v.x + v.y + v.z + v.w; }
        else if (part == 1) { acc[2 * rb]     += v.x + v.y;
                              acc[2 * rb + 1] += v.z + v.w; }
        else                { acc[2 * rb + 1] += v.x + v.y + v.z + v.w; }
    }
    float cw[4], sw[4];
#pragma unroll
    for (int w = 0; w < 4; ++w) {
        float a = acc[w] * (1.0f / 36.0f) * 0.5f;
        cw[w] = cosf(a); sw[w] = sinf(a);
    }

    // ---- encoded product state: amp[i] = (-i)^popc(i) * prod(c/s)
    const int col = wave * 32 + lane;
#pragma unroll
    for (int i = 0; i < DIM; ++i) {
        float m = ((i & 8) ? sw[0] : cw[0]) * ((i & 4) ? sw[1] : cw[1]) *
                  ((i & 2) ? sw[2] : cw[2]) * ((i & 1) ? sw[3] : cw[3]);
        int pc = __popc(i) & 3;
        float re = (pc == 0) ? m : ((pc == 2) ? -m : 0.0f);
        float im = (pc == 1) ? -m : ((pc == 3) ? m : 0.0f);
        PsiRe[i][col] = re;
        PsiIm[i][col] = im;
    }
    __syncthreads();

    // ---- A fragments of U (16x4 f32: lanes16-31 hold K+2,K+3)
    v2f ar[4], ai[4], an[4];
#pragma unroll
    for (int q = 0; q < 4; ++q) {
        int k = 4 * q + 2 * hi;
        ar[q] = (v2f){Ur[row * 16 + k], Ur[row * 16 + k + 1]};
        ai[q] = (v2f){Ui[row * 16 + k], Ui[row * 16 + k + 1]};
        an[q] = -ai[q];   // f32 WMMA has no A-negate modifier -> prenegate
    }

    float qv[4] = {0.f, 0.f, 0.f, 0.f};
#pragma unroll
    for (int tile = 0; tile < 2; ++tile) {
        const int bc = wave * 32 + tile * 16 + row;   // B col (N = lane&15)
        v8f aRe = {0.f, 0.f, 0.f, 0.f, 0.f, 0.f, 0.f, 0.f};
        v8f aIm = {0.f, 0.f, 0.f, 0.f, 0.f, 0.f, 0.f, 0.f};
#pragma unroll
        for (int q = 0; q < 4; ++q) {
            int kb = 4 * q + 2 * hi;                  // B 4x16: reg j -> row 2*hi+j
            v2f bre = (v2f){PsiRe[kb][bc], PsiRe[kb + 1][bc]};
            v2f bim = (v2f){PsiIm[kb][bc], PsiIm[kb + 1][bc]};
            // Re = Ur*Re - Ui*Im ; Im = Ur*Im + Ui*Re
            aRe = __builtin_amdgcn_wmma_f32_16x16x4_f32(false, ar[q], false, bre,
                                                        (short)0, aRe, false, false);
            aRe = __builtin_amdgcn_wmma_f32_16x16x4_f32(false, an[q], false, bim,
                                                        (short)0, aRe, false, false);
            aIm = __builtin_amdgcn_wmma_f32_16x16x4_f32(false, ar[q], false, bim,
                                                        (short)0, aIm, false, false);
            aIm = __builtin_amdgcn_wmma_f32_16x16x4_f32(false, ai[q], false, bre,
                                                        (short)0, aIm, false, false);
        }
        // probs + Z expectations (D rows g / g+8 across lane halves)
        float z[4] = {0.f, 0.f, 0.f, 0.f};
#pragma unroll
        for (int g = 0; g < 8; ++g) {
            float p = aRe[g] * aRe[g] + aIm[g] * aIm[g];
            int i = hi * 8 + g;
            z[0] += (i & 8) ? -p : p;
            z[1] += (i & 4) ? -p : p;
            z[2] += (i & 2) ? -p : p;
            z[3] += (i & 1) ? -p : p;
        }
#pragma unroll
        for (int w = 0; w < 4; ++w) z[w] = xor16_add(z[w]);
        if (tile == hi) {                 // lane L keeps sample waveBase + L
#pragma unroll
            for (int w = 0; w < 4; ++w) qv[w] = z[w];
        }
    }

    // ---- store raw q (normalized later)
    float4 o; o.x = qv[0]; o.y = qv[1]; o.z = qv[2]; o.w = qv[3];
    ((float4*)out)[sample] = o;

    // ---- deterministic per-block reduction of sum / sumsq
#pragma unroll
    for (int w = 0; w < 4; ++w) {
        red[w][tid]     = qv[w];
        red[4 + w][tid] = qv[w] * qv[w];
    }
    __syncthreads();
    for (int st = 128; st > 0; st >>= 1) {
        if (tid < st) {
#pragma unroll
            for (int v = 0; v < 8; ++v) red[v][tid] += red[v][tid + st];
        }
        __syncthreads();
    }
    if (tid == 0) {
#pragma unroll
        for (int v = 0; v < 8; ++v) partials[blockIdx.x * 8 + v] = red[v][0];
    }
}

// ---------------------------------------------------------------------------
// Kernel 2: fold per-block partials -> scale/shift for batchnorm.
// ---------------------------------------------------------------------------
__global__ __launch_bounds__(256) void qcirc_reduce(const float* __restrict__ partials,
                                                    int nblocks, float invB,
                                                    const float* __restrict__ gamma,
                                                    const float* __restrict__ beta,
                                                    float* __restrict__ stats) {
    __shared__ float red[8][256];
    const int tid = threadIdx.x;
    float a[8] = {0.f, 0.f, 0.f, 0.f, 0.f, 0.f, 0.f, 0.f};
    for (int i = tid; i < nblocks; i += 256)
#pragma unroll
        for (int v = 0; v < 8; ++v) a[v] += partials[i * 8 + v];
#pragma unroll
    for (int v = 0; v < 8; ++v) red[v][tid] = a[v];
    __syncthreads();
    for (int st = 128; st > 0; st >>= 1) {
        if (tid < st) {
#pragma unroll
            for (int v = 0; v < 8; ++v) red[v][tid] += red[v][tid + st];
        }
        __syncthreads();
    }
    if (tid == 0) {
#pragma unroll
        for (int w = 0; w < 4; ++w) {
            float mean = red[w][0] * invB;
            float var  = red[4 + w][0] * invB - mean * mean;
            float inv  = rsqrtf(var + 1e-5f);
            float sc   = gamma[w] * inv;
            stats[w]     = sc;
            stats[4 + w] = beta[w] - mean * sc;
        }
    }
}

// ---------------------------------------------------------------------------
// Kernel 3: in-place affine normalize of d_out.
// ---------------------------------------------------------------------------
__global__ __launch_bounds__(256) void qcirc_norm(float* __restrict__ out,
                                                  const float* __restrict__ stats) {
    const long idx = (long)blockIdx.x * 256 + threadIdx.x;  // one float4 / thread
    float4 q = ((float4*)out)[idx];
    q.x = q.x * stats[0] + stats[4];
    q.y = q.y * stats[1] + stats[5];
    q.z = q.z * stats[2] + stats[6];
    q.w = q.w * stats[3] + stats[7];
    ((float4*)out)[idx] = q;
}

extern "C" void kernel_launch(void* const* d_in, const int* in_sizes, int n_in,
                              void* d_out, int out_size, void* d_ws, size_t ws_size,
                              hipStream_t stream) {
    const float* x     = (const float*)d_in[0];
    const float* W     = (const float*)d_in[1];
    const float* gamma = (const float*)d_in[2];
    const float* beta  = (const float*)d_in[3];
    float* out = (float*)d_out;

    float* ws       = (float*)d_ws;
    float* Ur       = ws;            // 256 floats
    float* Ui       = ws + 256;      // 256 floats
    float* partials = ws + 512;      // nblocks * 8 floats
    const int B       = in_sizes[0] / 144;   // 131072
    const int nblocks = B / 256;             // 512
    float* stats    = partials + nblocks * 8; // 8 floats

    build_unitary<<<1, 32, 0, stream>>>(W, Ur, Ui);
    qcirc_main<<<nblocks, 256, 0, stream>>>(x, Ur, Ui, out, partials);
    qcirc_reduce<<<1, 256, 0, stream>>>(partials, nblocks, 1.0f / (float)B,
                                        gamma, beta, stats);
    qcirc_norm<<<B / 256, 256, 0, stream>>>(out, stats);
}